// PPIModel_36910948942110
// MI455X (gfx1250) — compile-verified
//
#include <hip/hip_runtime.h>
#include <math.h>

#define N_NODES 50000
#define N_EDGES 1600000
#define N_RELS  8
#define N_BASES 4
#define H       64

typedef __attribute__((ext_vector_type(2))) float v2f;
typedef __attribute__((ext_vector_type(8))) float v8f;

// h[n,d] = features[n,0]*W_in[0,d] + features[n,1]*W_in[1,d] + b_in[d]
__global__ void embed_kernel(const float* __restrict__ features,
                             const float* __restrict__ W_in,
                             const float* __restrict__ b_in,
                             float* __restrict__ h) {
  int idx = blockIdx.x * blockDim.x + threadIdx.x;
  if (idx >= N_NODES * H) return;
  int n = idx >> 6, d = idx & 63;
  h[idx] = features[2 * n] * W_in[d] + features[2 * n + 1] * W_in[H + d] + b_in[d];
}

// W[r,i,o] = sum_b w_comp[r,b] * bases[b,i,o]
__global__ void basis_kernel(const float* __restrict__ w_comp,
                             const float* __restrict__ bases,
                             float* __restrict__ W) {
  int idx = blockIdx.x * blockDim.x + threadIdx.x;
  if (idx >= N_RELS * H * H) return;
  int r = idx >> 12, io = idx & 4095;
  float s = 0.f;
#pragma unroll
  for (int b = 0; b < N_BASES; ++b)
    s += w_comp[r * N_BASES + b] * bases[b * H * H + io];
  W[idx] = s;
}

// proj[r] = h(N x 64) @ W[r](64 x 64); one wave = one 16x16 f32 tile via
// V_WMMA_F32_16X16X4_F32. Block = 128 threads = 4 waves = 16 rows x 64 cols.
__global__ void proj_kernel(const float* __restrict__ h,
                            const float* __restrict__ W,
                            float* __restrict__ proj) {
  const int lane = threadIdx.x & 31;
  const int wave = threadIdx.x >> 5;
  const int half = lane >> 4;      // 0: lanes 0-15, 1: lanes 16-31
  const int lmod = lane & 15;
  const int m0   = blockIdx.x * 16;
  const int n0   = wave * 16;
  const int r    = blockIdx.y;
  const float* Wr = W + r * H * H;

  v8f c = {};
#pragma unroll
  for (int k0 = 0; k0 < H; k0 += 4) {
    const int kb = k0 + half * 2;                      // A: K pair per lane-half
    v2f a = *(const v2f*)(h + (m0 + lmod) * H + kb);   // rows M=lane%16
    v2f b;
    b.x = Wr[kb * H + n0 + lmod];                      // B: K rows, N=lane%16
    b.y = Wr[(kb + 1) * H + n0 + lmod];
    c = __builtin_amdgcn_wmma_f32_16x16x4_f32(false, a, false, b,
                                              (short)0, c, false, false);
  }
  float* out = proj + (size_t)r * N_NODES * H;
#pragma unroll
  for (int j = 0; j < 8; ++j) {                        // C/D: VGPR j -> M=j(+8)
    int m = m0 + j + half * 8;
    out[m * H + n0 + lmod] = c[j];
  }
}

// agg[dst] += proj[etype, src]; 16 threads/edge, b128 gather + 4 HW f32 atomics
__global__ void edge_kernel(const int* __restrict__ src,
                            const int* __restrict__ dst,
                            const int* __restrict__ etype,
                            const float* __restrict__ proj,
                            float* __restrict__ agg) {
  long long t = (long long)blockIdx.x * blockDim.x + threadIdx.x;
  if (t >= (long long)N_EDGES * 16) return;
  int e = (int)(t >> 4);
  int q = (int)(t & 15);
  int r = etype[e], s = src[e], d = dst[e];
  const float4 v = *(const float4*)(proj + ((size_t)r * N_NODES + s) * H + q * 4);
  float* a = agg + (size_t)d * H + q * 4;
  unsafeAtomicAdd(a + 0, v.x);
  unsafeAtomicAdd(a + 1, v.y);
  unsafeAtomicAdd(a + 2, v.z);
  unsafeAtomicAdd(a + 3, v.w);
}

// out = agg + h@loop_w + h_bias (WMMA), fused with dot(out, fc_W) reduction
__global__ void final_kernel(const float* __restrict__ h,
                             const float* __restrict__ loop_w,
                             const float* __restrict__ agg,
                             const float* __restrict__ h_bias,
                             const float* __restrict__ fc_W,
                             float* __restrict__ acc) {
  __shared__ float sdata[4];
  const int lane = threadIdx.x & 31;
  const int wave = threadIdx.x >> 5;
  const int half = lane >> 4;
  const int lmod = lane & 15;
  const int m0   = blockIdx.x * 16;
  const int n0   = wave * 16;

  v8f c = {};
#pragma unroll
  for (int k0 = 0; k0 < H; k0 += 4) {
    const int kb = k0 + half * 2;
    v2f a = *(const v2f*)(h + (m0 + lmod) * H + kb);
    v2f b;
    b.x = loop_w[kb * H + n0 + lmod];
    b.y = loop_w[(kb + 1) * H + n0 + lmod];
    c = __builtin_amdgcn_wmma_f32_16x16x4_f32(false, a, false, b,
                                              (short)0, c, false, false);
  }
  float partial = 0.f;
#pragma unroll
  for (int j = 0; j < 8; ++j) {
    int m = m0 + j + half * 8;
    int n = n0 + lmod;
    float o = c[j] + agg[m * H + n] + h_bias[n];
    partial += o * fc_W[m * H + n];   // fc_W flat index = node*H + dim
  }
  for (int off = 16; off > 0; off >>= 1)
    partial += __shfl_down(partial, off, 32);
  if (lane == 0) sdata[wave] = partial;
  __syncthreads();
  if (threadIdx.x == 0)
    unsafeAtomicAdd(acc, sdata[0] + sdata[1] + sdata[2] + sdata[3]);
}

__global__ void sigmoid_kernel(const float* __restrict__ acc,
                               const float* __restrict__ fc_b,
                               float* __restrict__ out) {
  float x = acc[0] + fc_b[0];
  out[0] = 1.f / (1.f + __expf(-x));
}

extern "C" void kernel_launch(void* const* d_in, const int* in_sizes, int n_in,
                              void* d_out, int out_size, void* d_ws, size_t ws_size,
                              hipStream_t stream) {
  const float* features = (const float*)d_in[0];
  const int*   src      = (const int*)d_in[1];
  const int*   dst      = (const int*)d_in[2];
  const int*   etype    = (const int*)d_in[3];
  const float* W_in     = (const float*)d_in[4];
  const float* b_in     = (const float*)d_in[5];
  const float* w_comp   = (const float*)d_in[6];
  const float* bases    = (const float*)d_in[7];
  const float* loop_w   = (const float*)d_in[8];
  const float* h_bias   = (const float*)d_in[9];
  const float* fc_W     = (const float*)d_in[10];
  const float* fc_b     = (const float*)d_in[11];

  // workspace carve-up (~128.2 MB total)
  char* ws = (char*)d_ws;
  size_t off = 0;
  float* h    = (float*)(ws + off); off += (size_t)N_NODES * H * 4;          // 12.8 MB
  float* W    = (float*)(ws + off); off += (size_t)N_RELS * H * H * 4;       // 128 KB
  float* agg  = (float*)(ws + off); off += (size_t)N_NODES * H * 4;          // 12.8 MB
  float* acc  = (float*)(ws + off); off += 256;
  float* proj = (float*)(ws + off); off += (size_t)N_RELS * N_NODES * H * 4; // 102.4 MB

  embed_kernel<<<(N_NODES * H + 255) / 256, 256, 0, stream>>>(features, W_in, b_in, h);
  basis_kernel<<<(N_RELS * H * H + 255) / 256, 256, 0, stream>>>(w_comp, bases, W);
  proj_kernel<<<dim3(N_NODES / 16, N_RELS), 128, 0, stream>>>(h, W, proj);

  hipMemsetAsync(agg, 0, (size_t)N_NODES * H * 4, stream);
  hipMemsetAsync(acc, 0, 4, stream);

  edge_kernel<<<(N_EDGES * 16) / 256, 256, 0, stream>>>(src, dst, etype, proj, agg);
  final_kernel<<<N_NODES / 16, 128, 0, stream>>>(h, loop_w, agg, h_bias, fc_W, acc);
  sigmoid_kernel<<<1, 1, 0, stream>>>(acc, fc_b, (float*)d_out);
}